// GMM_score_rescaled_15556371546670
// MI455X (gfx1250) — compile-verified
//
#include <hip/hip_runtime.h>

typedef __attribute__((ext_vector_type(2))) float v2f;
typedef __attribute__((ext_vector_type(8))) float v8f;

#define LOG2E        1.4426950408889634f
#define LN_SIGMA_MAX 3.2188758248682006f   // ln(25.0)

#define N_TOTAL        16384
#define B_TOTAL        16384
#define SEGS           4
#define SEG_LEN        (N_TOTAL / SEGS)      // 4096
#define CHUNK          2048                  // floats staged in LDS per step (8 KB)
#define CHUNKS_PER_SEG (SEG_LEN / CHUNK)     // 2
#define WAVES_PER_BLK  8
#define TPB            256

__device__ __forceinline__ float fexp2(float v) { return __builtin_amdgcn_exp2f(v); }

__device__ __forceinline__ float sigma2_of_t(float tj) {
    // sigma^2 = (exp(2 t ln S) - 1) / (2 ln S)
    return (fexp2(2.0f * tj * LN_SIGMA_MAX * LOG2E) - 1.0f) / (2.0f * LN_SIGMA_MAX);
}

// Pass 1: each wave owns 16 consecutive j, accumulates partial num/den over one
// i-segment using V_WMMA_F32_16X16X4_F32:
//   D(16x16) += A(16x4) * B(4x16)
//   A row0 = y_{i..i+3}, A row1 = 1, rows 2..15 = 0
//   B[k][n] = pdf(i+k, j_n)
// -> D[0][n] = partial num, D[1][n] = partial den.
__global__ void __launch_bounds__(TPB)
gmm_pass1(const float* __restrict__ x,
          const float* __restrict__ t,
          const float* __restrict__ y,
          float* __restrict__ ws)
{
    __shared__ float sy[CHUNK];

    const int tid   = threadIdx.x;
    const int lane  = tid & 31;          // wave32
    const int wave  = tid >> 5;
    const int m     = lane & 15;         // column n (and row m for A)
    const int kbase = (lane >> 4) << 1;  // lanes 0-15 -> K0,K1 ; lanes 16-31 -> K2,K3
    const int jtile = blockIdx.x * WAVES_PER_BLK + wave;
    const int j     = jtile * 16 + m;
    const int seg   = blockIdx.y;
    const int ibase = seg * SEG_LEN;

    const float xj  = x[j];
    const float s2  = sigma2_of_t(t[j]);
    const float a   = -0.5f * LOG2E / s2;          // pdf = exp2(a * d * d)
    const float arest = (m == 1) ? 1.0f : 0.0f;    // A rows >= 1 pattern

    v8f acc = {};

    for (int ck = 0; ck < CHUNKS_PER_SEG; ++ck) {
        // Cooperative stage of CHUNK train values into LDS (uniform trip count).
        const float4* src = (const float4*)(y + ibase + ck * CHUNK);
        float4*       dst = (float4*)sy;
        #pragma unroll
        for (int v = 0; v < CHUNK / 4 / TPB; ++v)
            dst[tid + v * TPB] = src[tid + v * TPB];
        __syncthreads();

        #pragma unroll 4
        for (int c = 0; c < CHUNK; c += 4) {
            const float y0 = sy[c + kbase];
            const float y1 = sy[c + kbase + 1];
            const float d0 = y0 - xj;
            const float d1 = y1 - xj;
            const float p0 = fexp2(a * d0 * d0);
            const float p1 = fexp2(a * d1 * d1);

            v2f A; A.x = (m == 0) ? y0 : arest;   // A[m][kbase]
                   A.y = (m == 0) ? y1 : arest;   // A[m][kbase+1]
            v2f B; B.x = p0;                      // B[kbase][n]
                   B.y = p1;                      // B[kbase+1][n]

            acc = __builtin_amdgcn_wmma_f32_16x16x4_f32(
                false, A, false, B, (short)0, acc, false, false);
        }
        __syncthreads();
    }

    // D layout: VGPR0 lanes 0-15 = (M=0, N=lane) -> num ; VGPR1 = (M=1) -> den.
    if (lane < 16) {
        ws[seg * (2 * B_TOTAL) + j]            = acc[0];  // partial num
        ws[seg * (2 * B_TOTAL) + B_TOTAL + j]  = acc[1];  // partial den
    }
}

// Pass 2: deterministically combine the 4 segment partials and finalize.
__global__ void __launch_bounds__(TPB)
gmm_pass2(const float* __restrict__ x,
          const float* __restrict__ t,
          const float* __restrict__ ws,
          float* __restrict__ out)
{
    const int j = blockIdx.x * TPB + threadIdx.x;
    float num = 0.0f, den = 0.0f;
    #pragma unroll
    for (int s = 0; s < SEGS; ++s) {
        num += ws[s * (2 * B_TOTAL) + j];
        den += ws[s * (2 * B_TOTAL) + B_TOTAL + j];
    }
    const float xj = x[j];
    const float s2 = sigma2_of_t(t[j]);
    const float evals = (den == 0.0f) ? 0.0f : (num / den);
    out[j] = (evals - xj) / s2;
}

extern "C" void kernel_launch(void* const* d_in, const int* in_sizes, int n_in,
                              void* d_out, int out_size, void* d_ws, size_t ws_size,
                              hipStream_t stream) {
    const float* x = (const float*)d_in[0];  // (16384, 1)
    const float* t = (const float*)d_in[1];  // (16384,)
    const float* y = (const float*)d_in[2];  // (16384, 1) train_data
    float* out = (float*)d_out;              // (16384, 1)
    float* ws  = (float*)d_ws;               // SEGS * 2 * 16384 floats = 512 KB

    dim3 grid1(B_TOTAL / (16 * WAVES_PER_BLK), SEGS);  // (128, 4)
    gmm_pass1<<<grid1, TPB, 0, stream>>>(x, t, y, ws);
    gmm_pass2<<<B_TOTAL / TPB, TPB, 0, stream>>>(x, t, ws, out);
}